// GNN_11647951307108
// MI455X (gfx1250) — compile-verified
//
#include <hip/hip_runtime.h>
#include <cstdint>
#include <cstddef>

#define DFEAT 128
#define NEG_SLOPE 0.2f
#define EPS_F 1e-16f

typedef __attribute__((ext_vector_type(2))) float v2f;
typedef __attribute__((ext_vector_type(8))) float v8f;

// ---- order-preserving float<->uint encoding for atomicMax on floats ----
__device__ __forceinline__ unsigned f2ord(float f) {
  unsigned u = __float_as_uint(f);
  return (u & 0x80000000u) ? ~u : (u | 0x80000000u);
}
__device__ __forceinline__ float ord2f(unsigned u) {
  return (u & 0x80000000u) ? __uint_as_float(u & 0x7FFFFFFFu)
                           : __uint_as_float(~u);
}
// host-side constant: f2ord(-inf) = ~0xFF800000 = 0x007FFFFF
#define ORD_NEG_INF 0x007FFFFFu

// ---------------------------------------------------------------- fills
__global__ void gatv2_fill_f32(float* __restrict__ p, float v, int n) {
  int i = blockIdx.x * blockDim.x + threadIdx.x;
  if (i < n) p[i] = v;
}
__global__ void gatv2_fill_u32(unsigned* __restrict__ p, unsigned v, int n) {
  int i = blockIdx.x * blockDim.x + threadIdx.x;
  if (i < n) p[i] = v;
}

// ------------------------------------------------- GEMM: Y = X*W + b (row-major)
// X:[nrows x 128], W:[128 x 128], Y:[nrows x 128]
// Block = 256 threads = 8 waves. Block owns a 16-row strip; wave w owns cols
// [16w,16w+16). K-loop uses V_WMMA_F32_16X16X4_F32 (fp32, K=4 per issue).
__global__ __launch_bounds__(256) void gatv2_gemm128_wmma(
    const float* __restrict__ X, const float* __restrict__ W,
    const float* __restrict__ bias, float* __restrict__ Y, int nrows)
{
  __shared__ float sX[16 * DFEAT]; // 8 KB strip of X, shared by 8 waves

  const int wave = threadIdx.x >> 5;
  const int lane = threadIdx.x & 31;
  const int r0   = blockIdx.x * 16;

  for (int i = threadIdx.x; i < 16 * DFEAT; i += 256) {
    int row = r0 + (i >> 7);
    sX[i] = (row < nrows) ? X[(size_t)row * DFEAT + (i & (DFEAT - 1))] : 0.0f;
  }
  __syncthreads();

  const int c0   = wave * 16;
  const int half = lane >> 4;   // 0: lanes 0-15, 1: lanes 16-31
  const int mn   = lane & 15;   // M for A-frag, N for B-frag
  const int ks   = half * 2;    // K sub-offset per ISA 16x4 f32 layout

  v8f acc = {};
  #pragma unroll
  for (int k = 0; k < DFEAT; k += 4) {
    v2f a, b;
    // A 16x4: lanes0-15 hold K={k,k+1}, lanes16-31 hold K={k+2,k+3}
    a.x = sX[mn * DFEAT + k + ks];
    a.y = sX[mn * DFEAT + k + ks + 1];
    // B 4x16 mirrored layout; W is row-major [K][N]
    b.x = W[(size_t)(k + ks) * DFEAT + c0 + mn];
    b.y = W[(size_t)(k + ks + 1) * DFEAT + c0 + mn];
    acc = __builtin_amdgcn_wmma_f32_16x16x4_f32(
        /*neg_a=*/false, a, /*neg_b=*/false, b,
        /*c_mod=*/(short)0, acc, /*reuse_a=*/false, /*reuse_b=*/false);
  }

  const float bv = bias[c0 + mn];
  #pragma unroll
  for (int r = 0; r < 8; ++r) {
    int row = r0 + r + 8 * half; // C/D layout: VGPR r = M{r, r+8}
    if (row < nrows) Y[(size_t)row * DFEAT + c0 + mn] = acc[r] + bv;
  }
}

// ---------------------------------- per-edge logits + segment max (wave/edge)
__global__ __launch_bounds__(256) void gatv2_edge_logits_max(
    const float* __restrict__ xl, const float* __restrict__ xr,
    const int* __restrict__ src, const int* __restrict__ dst,
    const float* __restrict__ att, float* __restrict__ logits,
    unsigned* __restrict__ segmax, int E)
{
  int e    = blockIdx.x * 8 + (threadIdx.x >> 5);
  int lane = threadIdx.x & 31;
  if (e >= E) return;
  int s = src[e], d = dst[e];
  const float* pl = xl + (size_t)s * DFEAT;
  const float* pr = xr + (size_t)d * DFEAT;

  float part = 0.0f;
  #pragma unroll
  for (int t = 0; t < 4; ++t) {
    int idx = lane + 32 * t;
    float v = pl[idx] + pr[idx];
    v = (v > 0.0f) ? v : NEG_SLOPE * v;   // leaky_relu
    part += v * att[idx];
  }
  #pragma unroll
  for (int off = 16; off > 0; off >>= 1)
    part += __shfl_xor(part, off, 32);

  if (lane == 0) {
    logits[e] = part;
    atomicMax(&segmax[d], f2ord(part));
  }
}

// ------------------------------------- exp(logit - max) + segment sum (thread/edge)
__global__ void gatv2_edge_exp_sum(
    float* __restrict__ logits, const int* __restrict__ dst,
    const unsigned* __restrict__ segmax, float* __restrict__ den, int E)
{
  int e = blockIdx.x * blockDim.x + threadIdx.x;
  if (e >= E) return;
  int d = dst[e];
  float ex = __expf(logits[e] - ord2f(segmax[d]));
  logits[e] = ex; // in place: logits -> ex
  atomicAdd(&den[d], ex);
}

// ------------------------------------------- weighted aggregation (wave/edge)
__global__ __launch_bounds__(256) void gatv2_edge_aggregate(
    const float* __restrict__ xl, const float* __restrict__ ex,
    const int* __restrict__ src, const int* __restrict__ dst,
    const float* __restrict__ den, float* __restrict__ agg, int E)
{
  int e    = blockIdx.x * 8 + (threadIdx.x >> 5);
  int lane = threadIdx.x & 31;
  if (e >= E) return;
  int s = src[e], d = dst[e];
  float alpha = ex[e] / (den[d] + EPS_F);
  const float* pj = xl + (size_t)s * DFEAT;
  float* po = agg + (size_t)d * DFEAT;
  #pragma unroll
  for (int t = 0; t < 4; ++t) {
    int idx = lane + 32 * t;
    atomicAdd(&po[idx], alpha * pj[idx]);
  }
}

// ---------------------------------------------------- finalize: +bias (+relu)
__global__ void gatv2_bias_act(float* __restrict__ y, const float* __restrict__ bias,
                               int total, int do_relu)
{
  int i = blockIdx.x * blockDim.x + threadIdx.x;
  if (i >= total) return;
  float v = y[i] + bias[i & (DFEAT - 1)];
  if (do_relu) v = fmaxf(v, 0.0f);
  y[i] = v;
}

// ===========================================================================
extern "C" void kernel_launch(void* const* d_in, const int* in_sizes, int n_in,
                              void* d_out, int out_size, void* d_ws, size_t ws_size,
                              hipStream_t stream)
{
  (void)n_in; (void)out_size; (void)ws_size;
  const float* x     = (const float*)d_in[0];
  const int*   ei    = (const int*)  d_in[1];
  const float* Wl1   = (const float*)d_in[2];
  const float* Wr1   = (const float*)d_in[3];
  const float* att1  = (const float*)d_in[4];
  const float* Wl2   = (const float*)d_in[5];
  const float* Wr2   = (const float*)d_in[6];
  const float* att2  = (const float*)d_in[7];
  const float* bl1   = (const float*)d_in[8];
  const float* br1   = (const float*)d_in[9];
  const float* bias1 = (const float*)d_in[10];
  const float* bl2   = (const float*)d_in[11];
  const float* br2   = (const float*)d_in[12];
  const float* bias2 = (const float*)d_in[13];

  const int N = in_sizes[0] / DFEAT;
  const int E = in_sizes[1] / 2;
  const int* src = ei;
  const int* dst = ei + E;

  float* ws = (float*)d_ws;
  const size_t nd = (size_t)N * DFEAT;
  float*    xl     = ws;                         // [N,D]
  float*    xr     = ws + nd;                    // [N,D]
  float*    h      = ws + 2 * nd;                // agg1 -> relu'd h  [N,D]
  float*    elog   = ws + 3 * nd;                // logits -> ex      [E]
  unsigned* segmax = (unsigned*)(ws + 3 * nd + (size_t)E); // [N]
  float*    den    = (float*)(segmax + N);       // [N]
  float*    out    = (float*)d_out;              // agg2 + bias2      [N,D]

  dim3 b256(256);
  const int gRows = (N + 15) / 16;
  const int gEw   = (E + 7) / 8;          // wave-per-edge
  const int gE    = (E + 255) / 256;
  const int gN    = (N + 255) / 256;
  const int gND   = (int)((nd + 255) / 256);

  // ----------------- layer 1 -----------------
  gatv2_gemm128_wmma<<<gRows, b256, 0, stream>>>(x, Wl1, bl1, xl, N);
  gatv2_gemm128_wmma<<<gRows, b256, 0, stream>>>(x, Wr1, br1, xr, N);
  gatv2_fill_u32<<<gN, b256, 0, stream>>>(segmax, ORD_NEG_INF, N);
  gatv2_fill_f32<<<gN, b256, 0, stream>>>(den, 0.0f, N);
  gatv2_fill_f32<<<gND, b256, 0, stream>>>(h, 0.0f, (int)nd);
  gatv2_edge_logits_max<<<gEw, b256, 0, stream>>>(xl, xr, src, dst, att1, elog, segmax, E);
  gatv2_edge_exp_sum<<<gE, b256, 0, stream>>>(elog, dst, segmax, den, E);
  gatv2_edge_aggregate<<<gEw, b256, 0, stream>>>(xl, elog, src, dst, den, h, E);
  gatv2_bias_act<<<gND, b256, 0, stream>>>(h, bias1, (int)nd, /*relu=*/1);

  // ----------------- layer 2 -----------------
  gatv2_gemm128_wmma<<<gRows, b256, 0, stream>>>(h, Wl2, bl2, xl, N);
  gatv2_gemm128_wmma<<<gRows, b256, 0, stream>>>(h, Wr2, br2, xr, N);
  gatv2_fill_u32<<<gN, b256, 0, stream>>>(segmax, ORD_NEG_INF, N);
  gatv2_fill_f32<<<gN, b256, 0, stream>>>(den, 0.0f, N);
  gatv2_fill_f32<<<gND, b256, 0, stream>>>(out, 0.0f, (int)nd);
  gatv2_edge_logits_max<<<gEw, b256, 0, stream>>>(xl, xr, src, dst, att2, elog, segmax, E);
  gatv2_edge_exp_sum<<<gE, b256, 0, stream>>>(elog, dst, segmax, den, E);
  gatv2_edge_aggregate<<<gEw, b256, 0, stream>>>(xl, elog, src, dst, den, out, E);
  gatv2_bias_act<<<gND, b256, 0, stream>>>(out, bias2, (int)nd, /*relu=*/0);
}